// SelectiveSSM_51110110822784
// MI455X (gfx1250) — compile-verified
//
#include <hip/hip_runtime.h>
#include <stdint.h>

typedef __attribute__((ext_vector_type(2))) float v2f;
typedef __attribute__((ext_vector_type(4))) float v4f;
typedef __attribute__((ext_vector_type(8))) float v8f;

#define HIDDEN 1024
#define STATE  16
#define BATCH  4
#define SEQ    2048
#define MROWS  (BATCH * SEQ)          // 8192
#define NPROJ  (2 * STATE + HIDDEN)   // 1056
#define XPLD   1152                   // padded ld for xp (9 * 128)
#define KT     16                     // K chunk staged in LDS
#define ROWPAD 20                     // LDS row pitch in floats (16 data + 4 pad)
#define NSTAGE 3                      // async pipeline depth

// ---------------------------------------------------------------------------
// CDNA5 async memory->LDS helpers (CDNA5 ISA / 15.18, ASYNCcnt).
// ASYNCcnt completes in order, so waiting "<= 4" drains exactly the oldest
// staged chunk (4 b128 per thread) while the next chunk stays in flight.
// ---------------------------------------------------------------------------
__device__ __forceinline__ uint32_t lds_offset(const void* p) {
    // flat shared-aperture pointers carry the LDS byte offset in addr[31:0]
    return (uint32_t)(uintptr_t)p;
}

__device__ __forceinline__ void async_cp16(uint32_t lds_addr, const float* gsrc) {
    asm volatile("global_load_async_to_lds_b128 %0, %1, off"
                 :: "v"(lds_addr), "v"((uint64_t)(uintptr_t)gsrc)
                 : "memory");
}

__device__ __forceinline__ void wait_async0() {
    asm volatile("s_wait_asynccnt 0" ::: "memory");
}
__device__ __forceinline__ void wait_async4() {
    asm volatile("s_wait_asynccnt 4" ::: "memory");
}

// ---------------------------------------------------------------------------
// C[m,n] = sum_k A[m,k] * W[n,k]  (A @ W.T), fp32 WMMA 16x16x4.
// Block: 256 threads (8 waves), tile 128(M) x 128(N), K chunked by 16,
// triple-buffered async-LDS staging. Wave (w&3, w>>2) owns 32M x 64N:
// 2 M-frags x 4 N-frags = 8 accumulators; per k-step: 6 ds_load_b64 -> 8 wmma.
// W row indices are clamped to N-1 so a padded-N grid stays uniform.
// ---------------------------------------------------------------------------
__global__ __launch_bounds__(256)
void gemm_wmma_f32(const float* __restrict__ A,
                   const float* __restrict__ W,
                   float* __restrict__ C,
                   int M, int N, int K, int ldc) {
    __shared__ float sx[NSTAGE][128 * ROWPAD];
    __shared__ float sw[NSTAGE][128 * ROWPAD];

    const int tid  = threadIdx.x;
    const int lane = tid & 31;
    const int wave = tid >> 5;
    const int half = lane >> 4;       // K offset 0,1 vs 2,3
    const int lr   = lane & 15;
    const int koff = half * 2;

    const int m0b = blockIdx.x * 128;
    const int n0b = blockIdx.y * 128;
    const int mf  = (wave & 3) * 32;  // wave M offset in tile
    const int nf  = (wave >> 2) * 64; // wave N offset in tile

    // staging addressing: 4 threads per row (16B each), 64 rows per pass
    const int skv = (tid & 3) * 4;
    const int sr  = tid >> 2;
    const float* gx0 = A + (size_t)(m0b + sr) * K + skv;
    const float* gx1 = A + (size_t)(m0b + sr + 64) * K + skv;
    int wr0 = n0b + sr;      if (wr0 > N - 1) wr0 = N - 1;
    int wr1 = n0b + sr + 64; if (wr1 > N - 1) wr1 = N - 1;
    const float* gw0 = W + (size_t)wr0 * K + skv;
    const float* gw1 = W + (size_t)wr1 * K + skv;

    const uint32_t dx0 = lds_offset(&sx[0][sr * ROWPAD + skv]);
    const uint32_t dx1 = lds_offset(&sx[0][(sr + 64) * ROWPAD + skv]);
    const uint32_t dw0 = lds_offset(&sw[0][sr * ROWPAD + skv]);
    const uint32_t dw1 = lds_offset(&sw[0][(sr + 64) * ROWPAD + skv]);
    const uint32_t bufstep = (uint32_t)(128 * ROWPAD * sizeof(float));

    v8f acc[2][4];
#pragma unroll
    for (int m = 0; m < 2; ++m)
#pragma unroll
        for (int i = 0; i < 4; ++i) acc[m][i] = (v8f){};

    const int nch = K / KT;

    auto stage = [&](int c, int b) {
        const int k0 = c * KT;
        const uint32_t bo = (uint32_t)b * bufstep;
        async_cp16(dx0 + bo, gx0 + k0);
        async_cp16(dx1 + bo, gx1 + k0);
        async_cp16(dw0 + bo, gw0 + k0);
        async_cp16(dw1 + bo, gw1 + k0);
    };

    auto compute = [&](int cur) {
#pragma unroll
        for (int ks = 0; ks < KT / 4; ++ks) {
            const int kk = ks * 4 + koff;
            v2f a0 = *(const v2f*)&sx[cur][(mf + lr) * ROWPAD + kk];
            v2f a1 = *(const v2f*)&sx[cur][(mf + 16 + lr) * ROWPAD + kk];
#pragma unroll
            for (int i = 0; i < 4; ++i) {
                v2f b = *(const v2f*)&sw[cur][(nf + i * 16 + lr) * ROWPAD + kk];
                acc[0][i] = __builtin_amdgcn_wmma_f32_16x16x4_f32(
                    false, a0, false, b, (short)0, acc[0][i], false, false);
                acc[1][i] = __builtin_amdgcn_wmma_f32_16x16x4_f32(
                    false, a1, false, b, (short)0, acc[1][i], false, false);
            }
        }
    };

    // 3-stage pipeline: two chunks in flight, wait only for the oldest.
    stage(0, 0);
    if (nch > 1) stage(1, 1);
    int cur = 0;
    for (int c = 0; c < nch - 1; ++c) {
        wait_async4();        // oldest chunk (4 async ops) complete, per-wave
        __syncthreads();      // barrier publishes all waves' LDS writes
        if (c + 2 < nch) {
            int nb = cur + 2; if (nb >= NSTAGE) nb -= NSTAGE;
            stage(c + 2, nb);
        }
        compute(cur);
        cur = (cur + 1 == NSTAGE) ? 0 : cur + 1;
    }
    wait_async0();
    __syncthreads();
    compute(cur);

#pragma unroll
    for (int m = 0; m < 2; ++m)
#pragma unroll
        for (int i = 0; i < 4; ++i) {
            const int col   = n0b + nf + i * 16 + lr;
            const int rbase = m0b + mf + m * 16 + half * 8;
#pragma unroll
            for (int j = 0; j < 8; ++j)
                C[(size_t)(rbase + j) * ldc + col] = acc[m][i][j];
        }
}

// ---------------------------------------------------------------------------
// Selective scan: one thread per (b,h) channel, 16 states in registers.
// ---------------------------------------------------------------------------
__global__ __launch_bounds__(256)
void ssm_scan(const float* __restrict__ xp,    // [B, SEQ, XPLD] (padded)
              const float* __restrict__ x,     // [B, SEQ, HIDDEN]
              const float* __restrict__ A_log, // [STATE]
              const float* __restrict__ Dv,    // [HIDDEN]
              float* __restrict__ y) {         // [B, SEQ, HIDDEN]
    const int b = blockIdx.x >> 2;
    const int h = ((blockIdx.x & 3) << 8) + threadIdx.x;

    float Ac[STATE];
#pragma unroll
    for (int n = 0; n < STATE; ++n) Ac[n] = -__expf(A_log[n]);
    const float dcoef = Dv[h];

    float hs[STATE];
#pragma unroll
    for (int n = 0; n < STATE; ++n) hs[n] = 0.0f;

    const float* xpb = xp + (size_t)b * SEQ * XPLD;
    const float* xb  = x  + (size_t)b * SEQ * HIDDEN + h;
    float*       yb  = y  + (size_t)b * SEQ * HIDDEN + h;

    for (int t = 0; t < SEQ; ++t) {
        const float* row = xpb + (size_t)t * XPLD;

        float Bv[STATE], Cv[STATE];
#pragma unroll
        for (int q = 0; q < 4; ++q) {
            v4f tb = *(const v4f*)(row + 4 * q);
            v4f tc = *(const v4f*)(row + STATE + 4 * q);
#pragma unroll
            for (int j = 0; j < 4; ++j) { Bv[4 * q + j] = tb[j]; Cv[4 * q + j] = tc[j]; }
        }

        const float dtr = row[2 * STATE + h];
        const float dt  = (dtr > 20.0f) ? dtr : __logf(1.0f + __expf(dtr));
        const float xt  = xb[(size_t)t * HIDDEN];

        float accy = dcoef * xt;
        const float dx = dt * xt;
#pragma unroll
        for (int n = 0; n < STATE; ++n) {
            const float dA = __expf(Ac[n] * dt);
            hs[n] = dA * hs[n] + Bv[n] * dx;
            accy += hs[n] * Cv[n];
        }
        yb[(size_t)t * HIDDEN] = accy;
    }
}

// ---------------------------------------------------------------------------
extern "C" void kernel_launch(void* const* d_in, const int* in_sizes, int n_in,
                              void* d_out, int out_size, void* d_ws, size_t ws_size,
                              hipStream_t stream) {
    const float* x     = (const float*)d_in[0]; // [4, 2048, 1024]
    const float* Wx    = (const float*)d_in[1]; // [1056, 1024]
    const float* A_log = (const float*)d_in[2]; // [16]
    const float* Dv    = (const float*)d_in[3]; // [1024]
    const float* Wout  = (const float*)d_in[4]; // [1024, 1024]
    float* out = (float*)d_out;                 // [4, 2048, 1024]

    float* xp = (float*)d_ws;                                   // 8192 x 1152 f32
    float* y  = (float*)((char*)d_ws +
                         (size_t)MROWS * XPLD * sizeof(float)); // 8192 x 1024 f32

    dim3 blk(256);

    // xp = x @ Wx.T : M=8192, N=1056 (padded grid to 1152), K=1024
    gemm_wmma_f32<<<dim3(MROWS / 128, XPLD / 128), blk, 0, stream>>>(
        x, Wx, xp, MROWS, NPROJ, HIDDEN, XPLD);

    // recurrent scan -> y
    ssm_scan<<<dim3((BATCH * HIDDEN) / 256), blk, 0, stream>>>(
        xp, x, A_log, Dv, y);

    // out = y @ Wout.T : M=8192, N=1024, K=1024
    gemm_wmma_f32<<<dim3(MROWS / 128, HIDDEN / 128), blk, 0, stream>>>(
        y, Wout, out, MROWS, HIDDEN, HIDDEN, HIDDEN);
}